// MarkovGPPrior_2199023256223
// MI455X (gfx1250) — compile-verified
//
#include <hip/hip_runtime.h>
#include <math.h>

#define NSTEPS 131072
#define CHUNK_L 128
#define NCHUNK (NSTEPS / CHUNK_L) /* 1024 */

typedef __attribute__((ext_vector_type(2))) float v2f;
typedef __attribute__((ext_vector_type(8))) float v8f;
typedef int v4i_async __attribute__((__vector_size__(4 * sizeof(int))));

union U8 { v8f v; float f[8]; };
union U2 { v2f v; float f[2]; };

#if defined(__HIP_DEVICE_COMPILE__) && __has_builtin(__builtin_amdgcn_global_load_async_to_lds_b128)
#define KF_ASYNC 1
#else
#define KF_ASYNC 0
#endif

#if defined(__HIP_DEVICE_COMPILE__) && __has_builtin(__builtin_amdgcn_s_wait_asynccnt)
#define KF_WAIT_ASYNC(n) __builtin_amdgcn_s_wait_asynccnt(n)
#else
#define KF_WAIT_ASYNC(n) asm volatile("s_wait_asynccnt %0" ::"i"(n) : "memory")
#endif

__device__ __forceinline__ float shfl32(float x, int s) { return __shfl(x, s, 32); }

__device__ __forceinline__ v8f wmma4(v2f a, v2f b, v8f c) {
  // D(16x16,f32) = A(16x4,f32) * B(4x16,f32) + C
  return __builtin_amdgcn_wmma_f32_16x16x4_f32(false, a, false, b, (short)0, c,
                                               false, false);
}

// ---------- small scalar 4x4 helpers ----------
__device__ __forceinline__ void mat4_mul(const float* A, const float* B, float* C) {
#pragma unroll
  for (int i = 0; i < 4; ++i)
#pragma unroll
    for (int j = 0; j < 4; ++j) {
      float s = 0.f;
#pragma unroll
      for (int k = 0; k < 4; ++k) s += A[i * 4 + k] * B[k * 4 + j];
      C[i * 4 + j] = s;
    }
}
__device__ __forceinline__ void mat4_mul_nt(const float* A, const float* B, float* C) {
  // C = A * B^T
#pragma unroll
  for (int i = 0; i < 4; ++i)
#pragma unroll
    for (int j = 0; j < 4; ++j) {
      float s = 0.f;
#pragma unroll
      for (int k = 0; k < 4; ++k) s += A[i * 4 + k] * B[j * 4 + k];
      C[i * 4 + j] = s;
    }
}
__device__ __forceinline__ void mat4_vec(const float* A, const float* x, float* y) {
#pragma unroll
  for (int i = 0; i < 4; ++i) {
    float s = 0.f;
#pragma unroll
    for (int k = 0; k < 4; ++k) s += A[i * 4 + k] * x[k];
    y[i] = s;
  }
}
__device__ void inv4(const float* a, float* o) {
  float s0 = a[0] * a[5] - a[4] * a[1];
  float s1 = a[0] * a[6] - a[4] * a[2];
  float s2 = a[0] * a[7] - a[4] * a[3];
  float s3 = a[1] * a[6] - a[5] * a[2];
  float s4 = a[1] * a[7] - a[5] * a[3];
  float s5 = a[2] * a[7] - a[6] * a[3];
  float c5 = a[10] * a[15] - a[14] * a[11];
  float c4 = a[9] * a[15] - a[13] * a[11];
  float c3 = a[9] * a[14] - a[13] * a[10];
  float c2 = a[8] * a[15] - a[12] * a[11];
  float c1 = a[8] * a[14] - a[12] * a[10];
  float c0 = a[8] * a[13] - a[12] * a[9];
  float det = s0 * c5 - s1 * c4 + s2 * c3 + s3 * c2 - s4 * c1 + s5 * c0;
  float id = 1.0f / det;
  o[0] = (a[5] * c5 - a[6] * c4 + a[7] * c3) * id;
  o[1] = (-a[1] * c5 + a[2] * c4 - a[3] * c3) * id;
  o[2] = (a[13] * s5 - a[14] * s4 + a[15] * s3) * id;
  o[3] = (-a[9] * s5 + a[10] * s4 - a[11] * s3) * id;
  o[4] = (-a[4] * c5 + a[6] * c2 - a[7] * c1) * id;
  o[5] = (a[0] * c5 - a[2] * c2 + a[3] * c1) * id;
  o[6] = (-a[12] * s5 + a[14] * s2 - a[15] * s1) * id;
  o[7] = (a[8] * s5 - a[10] * s2 + a[11] * s1) * id;
  o[8] = (a[4] * c4 - a[5] * c2 + a[7] * c0) * id;
  o[9] = (-a[0] * c4 + a[1] * c2 - a[3] * c0) * id;
  o[10] = (a[12] * s4 - a[13] * s2 + a[15] * s0) * id;
  o[11] = (-a[8] * s4 + a[9] * s2 - a[11] * s0) * id;
  o[12] = (-a[4] * c3 + a[5] * c1 - a[6] * c0) * id;
  o[13] = (a[0] * c3 - a[1] * c1 + a[2] * c0) * id;
  o[14] = (-a[12] * s3 + a[13] * s1 - a[14] * s0) * id;
  o[15] = (a[8] * s3 - a[9] * s1 + a[10] * s0) * id;
}

// ======================================================================
// Kernel 1: sequential Kalman filter, one wave32, lane (r,c) holds matrix
// element; all 4x4 algebra via cross-lane shuffles.
// ======================================================================
__global__ void kf_filter_kernel(const float* __restrict__ H,
                                 const float* __restrict__ Pinf,
                                 const float* __restrict__ A_seq,
                                 const float* __restrict__ Q_seq,
                                 const float* __restrict__ resid,
                                 const float* __restrict__ maskv,
                                 const float* __restrict__ R_seq,
                                 float* __restrict__ mp_ws, float* __restrict__ Pp_ws,
                                 float* __restrict__ mf_ws, float* __restrict__ Pf_ws,
                                 float* __restrict__ out) {
  const int lane = threadIdx.x & 31;
  const int r = (lane >> 2) & 3;
  const int c = lane & 3;
  const float log2pi = 1.8378770664093453f;
  const float hr = H[r];
  float m = 0.0f;
  float P = Pinf[r * 4 + c];
  float ll = 0.0f;

  for (int n = 0; n < NSTEPS; ++n) {
    if (n + 16 < NSTEPS) {
      __builtin_prefetch(&A_seq[(n + 16) * 16], 0, 1);
      __builtin_prefetch(&Q_seq[(n + 16) * 16], 0, 1);
    }
    float a = A_seq[n * 16 + r * 4 + c];
    float q = Q_seq[n * 16 + r * 4 + c];
    float rn = resid[n];
    float mk = maskv[n];
    float Rn = R_seq[n];

    float mp = 0.f;
#pragma unroll
    for (int k = 0; k < 4; ++k) mp += shfl32(a, 4 * r + k) * shfl32(m, 4 * k);
    float ap = 0.f;
#pragma unroll
    for (int k = 0; k < 4; ++k) ap += shfl32(a, 4 * r + k) * shfl32(P, 4 * k + c);
    float Pp = q;
#pragma unroll
    for (int k = 0; k < 4; ++k) Pp += shfl32(ap, 4 * r + k) * shfl32(a, 4 * c + k);
    Pp = 0.5f * (Pp + shfl32(Pp, 4 * c + r));

    float Ph = 0.f;
#pragma unroll
    for (int k = 0; k < 4; ++k) Ph += shfl32(Pp, 4 * r + k) * shfl32(hr, 4 * k);
    float S = Rn, hm = 0.f;
#pragma unroll
    for (int k = 0; k < 4; ++k) {
      S += shfl32(hr, 4 * k) * shfl32(Ph, 4 * k);
      hm += shfl32(hr, 4 * k) * shfl32(mp, 4 * k);
    }
    float innov = rn - hm;
    bool obs = (mk == 1.0f);
    float Ssafe = obs ? S : 1.0f;
    float K = obs ? (Ph / Ssafe) : 0.0f;
    float mnew = mp + K * innov;

    float ikh = ((r == c) ? 1.0f : 0.0f) - K * shfl32(hr, 4 * c);
    float t1 = 0.f;
#pragma unroll
    for (int k = 0; k < 4; ++k) t1 += shfl32(ikh, 4 * r + k) * shfl32(Pp, 4 * k + c);
    float Pn = Rn * K * shfl32(K, 4 * c);
#pragma unroll
    for (int k = 0; k < 4; ++k) Pn += shfl32(t1, 4 * r + k) * shfl32(ikh, 4 * c + k);
    Pn = 0.5f * (Pn + shfl32(Pn, 4 * c + r));

    if (obs) ll += -0.5f * (log2pi + logf(Ssafe) + innov * innov / Ssafe);

    if (lane < 16) {
      Pp_ws[n * 16 + lane] = Pp;
      Pf_ws[n * 16 + lane] = Pn;
      if (c == 0) {
        mp_ws[n * 4 + r] = mp;
        mf_ws[n * 4 + r] = mnew;
      }
    }
    m = mnew;
    P = Pn;
  }
  if (threadIdx.x == 0) out[2 * NSTEPS] = ll;
}

// ======================================================================
// Kernel 2: per-step smoother-element construction (fully parallel).
// ======================================================================
__global__ void kf_prep_kernel(const float* __restrict__ A_seq,
                               const float* __restrict__ mp_ws,
                               const float* __restrict__ Pp_ws,
                               const float* __restrict__ mf_ws,
                               const float* __restrict__ Pf_ws,
                               float* __restrict__ J_ws, float* __restrict__ c_ws,
                               float* __restrict__ E_ws) {
  int n = blockIdx.x * blockDim.x + threadIdx.x;
  if (n >= NSTEPS) return;
  if (n == NSTEPS - 1) {
#pragma unroll
    for (int i = 0; i < 16; ++i) {
      J_ws[(size_t)n * 16 + i] = ((i >> 2) == (i & 3)) ? 1.f : 0.f;
      E_ws[(size_t)n * 16 + i] = 0.f;
    }
#pragma unroll
    for (int i = 0; i < 4; ++i) c_ws[(size_t)n * 4 + i] = 0.f;
    return;
  }
  float Pf[16], A1[16], Pp1[16], Pinv[16], T[16], J[16], JP[16], JPJ[16];
  float mf[4], mp1[4], cv[4], tmp[4];
#pragma unroll
  for (int i = 0; i < 16; ++i) {
    Pf[i] = Pf_ws[(size_t)n * 16 + i];
    A1[i] = A_seq[(size_t)(n + 1) * 16 + i];
    Pp1[i] = Pp_ws[(size_t)(n + 1) * 16 + i];
  }
#pragma unroll
  for (int i = 0; i < 4; ++i) {
    mf[i] = mf_ws[(size_t)n * 4 + i];
    mp1[i] = mp_ws[(size_t)(n + 1) * 4 + i];
  }
  mat4_mul_nt(Pf, A1, T);
  inv4(Pp1, Pinv);
  mat4_mul(T, Pinv, J);
  mat4_vec(J, mp1, tmp);
#pragma unroll
  for (int i = 0; i < 4; ++i) cv[i] = mf[i] - tmp[i];
  mat4_mul(J, Pp1, JP);
  mat4_mul_nt(JP, J, JPJ);
#pragma unroll
  for (int i = 0; i < 16; ++i) E_ws[(size_t)n * 16 + i] = Pf[i] - JPJ[i];
#pragma unroll
  for (int i = 0; i < 16; ++i) J_ws[(size_t)n * 16 + i] = J[i];
#pragma unroll
  for (int i = 0; i < 4; ++i) c_ws[(size_t)n * 4 + i] = cv[i];
}

// ======================================================================
// Async staging of one scan step's (J, E, c) for 4 chunks into LDS.
// Lanes 0-15 fetch J rows (b128 each), lanes 16-31 fetch E rows,
// lanes 0-3 additionally fetch c. Tracked by ASYNCcnt.
// ======================================================================
__device__ __forceinline__ void stage_step(const float* __restrict__ Jg,
                                           const float* __restrict__ cg,
                                           const float* __restrict__ Eg,
                                           int blk4, int n, int lane,
                                           float* Jst, float* Est, float* cst) {
  int b = (lane >> 2) & 3;
  int p = lane & 3;
  size_t go = ((size_t)(blk4 + b) * CHUNK_L + n) * 16 + (size_t)p * 4;
  const float* gsrc = (lane < 16) ? (Jg + go) : (Eg + go);
  float* ldst = (lane < 16) ? (Jst + b * 16 + p * 4) : (Est + b * 16 + p * 4);
#if KF_ASYNC
  __builtin_amdgcn_global_load_async_to_lds_b128(
      (v4i_async*)gsrc,
      (__attribute__((address_space(3))) v4i_async*)ldst, 0, 0);
  if (lane < 4) {
    const float* csrc = cg + ((size_t)(blk4 + lane) * CHUNK_L + n) * 4;
    __builtin_amdgcn_global_load_async_to_lds_b128(
        (v4i_async*)csrc,
        (__attribute__((address_space(3))) v4i_async*)(cst + lane * 4), 0, 0);
  }
#else
#pragma unroll
  for (int i = 0; i < 4; ++i) ldst[i] = gsrc[i];
  if (lane < 4) {
    const float* csrc = cg + ((size_t)(blk4 + lane) * CHUNK_L + n) * 4;
    float* cd = cst + lane * 4;
#pragma unroll
    for (int i = 0; i < 4; ++i) cd[i] = csrc[i];
  }
#endif
}

// ======================================================================
// Kernel 3a: chunk summaries via WMMA with double-buffered async LDS
// staging. One wave per block processes 4 chunks packed into 16x16x4
// f32 WMMA ops; diagonal blocks of the 16x16 result are the 4 products.
// (M,G,c) <- (J M, (J G) J^T + E, J c + c_n), n descending.
// ======================================================================
__global__ void __launch_bounds__(32)
kf_summary_kernel(const float* __restrict__ J_ws, const float* __restrict__ c_ws,
                  const float* __restrict__ E_ws, float* __restrict__ Msum,
                  float* __restrict__ csum, float* __restrict__ Gsum) {
  __shared__ float Mbuf[4][16];  // canonical 4x16: Mbuf[r][4b+c] = M_b[r][c]
  __shared__ float Gbuf[4][16];
  __shared__ float Tbuf[4][16];
  __shared__ float cbuf[16];
  __shared__ float Jst[2][64];   // staged J, row-major per chunk
  __shared__ float Est[2][64];   // staged E
  __shared__ float cst[2][16];   // staged c

  const int lane = threadIdx.x & 31;
  const int blk4 = blockIdx.x * 4;
  const int ncol = (lane < 16) ? lane : (lane - 16);
  const int nb = ncol >> 2;              // block of this lane's column
  const int rA = ncol & 3;               // row within block (A layout)
  const int kOff = (lane < 16) ? 0 : 2;  // K columns 0,1 or 2,3

  for (int idx = lane; idx < 64; idx += 32) {
    int rr = idx >> 4, col = idx & 15;
    Mbuf[rr][col] = ((col & 3) == rr) ? 1.0f : 0.0f;
    Gbuf[rr][col] = 0.0f;
    Tbuf[rr][col] = 0.0f;
  }
  if (lane < 16) cbuf[lane] = 0.0f;

  // prologue: stage step L-1 into buffer 0
  stage_step(J_ws, c_ws, E_ws, blk4, CHUNK_L - 1, lane, Jst[0], Est[0], cst[0]);
  __syncthreads();

  const v8f zero8 = {0.f, 0.f, 0.f, 0.f, 0.f, 0.f, 0.f, 0.f};
  int cur = 0;

  for (int n = CHUNK_L - 1; n >= 0; --n) {
    // overlap: stage next step into the other buffer while computing
    if (n > 0) {
      stage_step(J_ws, c_ws, E_ws, blk4, n - 1, lane, Jst[1 - cur], Est[1 - cur],
                 cst[1 - cur]);
      KF_WAIT_ASYNC(2);  // drain the older pair -> buf[cur] ready
    } else {
      KF_WAIT_ASYNC(0);
    }
    __syncthreads();

    const float* Jc = Jst[cur];
    const float* Ec = Est[cur];
    const float* cc = cst[cur];

    // ---- A-layout J; identical elements serve as B-layout J^T ----
    U2 aJ;
    aJ.f[0] = Jc[nb * 16 + rA * 4 + kOff];
    aJ.f[1] = Jc[nb * 16 + rA * 4 + kOff + 1];
    // ---- B-layout running M, G from LDS ----
    U2 bM, bG;
    bM.f[0] = Mbuf[kOff][ncol];
    bM.f[1] = Mbuf[kOff + 1][ncol];
    bG.f[0] = Gbuf[kOff][ncol];
    bG.f[1] = Gbuf[kOff + 1][ncol];

    U8 Dm, Dg;
    Dm.v = wmma4(aJ.v, bM.v, zero8);  // J * M
    Dg.v = wmma4(aJ.v, bG.v, zero8);  // J * G
    __syncthreads();
    // ---- extract diagonal blocks back to canonical LDS ----
    if (lane < 16) {
      if (nb < 2) {
#pragma unroll
        for (int rr = 0; rr < 4; ++rr) {
          Mbuf[rr][ncol] = Dm.f[4 * nb + rr];
          Tbuf[rr][ncol] = Dg.f[4 * nb + rr];
        }
      }
    } else {
      if (nb >= 2) {
#pragma unroll
        for (int rr = 0; rr < 4; ++rr) {
          Mbuf[rr][ncol] = Dm.f[4 * nb + rr - 8];
          Tbuf[rr][ncol] = Dg.f[4 * nb + rr - 8];
        }
      }
    }
    __syncthreads();

    // ---- G' = (J G) * J^T + E (E packed block-diagonal into C) ----
    U2 aT;
    aT.f[0] = Tbuf[rA][4 * nb + kOff];
    aT.f[1] = Tbuf[rA][4 * nb + kOff + 1];
    U8 Ce;
#pragma unroll
    for (int v = 0; v < 8; ++v) {
      int mrow = (lane < 16) ? v : (v + 8);
      Ce.f[v] = ((mrow >> 2) == nb) ? Ec[nb * 16 + (mrow - 4 * nb) * 4 + rA] : 0.0f;
    }
    U8 DG;
    DG.v = wmma4(aT.v, aJ.v, Ce.v);  // B operand == A-layout J (same elements)

    // ---- c' = J c + c_n ----
    float cnew = 0.f;
    if (lane < 16) {
      cnew = cc[lane];
#pragma unroll
      for (int k = 0; k < 4; ++k)
        cnew += Jc[(lane >> 2) * 16 + (lane & 3) * 4 + k] * cbuf[(lane & ~3) + k];
    }
    __syncthreads();
    if (lane < 16) {
      if (nb < 2) {
#pragma unroll
        for (int rr = 0; rr < 4; ++rr) Gbuf[rr][ncol] = DG.f[4 * nb + rr];
      }
      cbuf[lane] = cnew;
    } else {
      if (nb >= 2) {
#pragma unroll
        for (int rr = 0; rr < 4; ++rr) Gbuf[rr][ncol] = DG.f[4 * nb + rr - 8];
      }
    }
    __syncthreads();
    cur ^= 1;
  }

  for (int idx = lane; idx < 64; idx += 32) {
    int b = idx >> 4, rc = idx & 15, rr = rc >> 2, cc2 = rc & 3;
    Msum[(size_t)(blk4 + b) * 16 + rc] = Mbuf[rr][4 * b + cc2];
    Gsum[(size_t)(blk4 + b) * 16 + rc] = Gbuf[rr][4 * b + cc2];
  }
  if (lane < 16) csum[(size_t)(blk4 + (lane >> 2)) * 4 + (lane & 3)] = cbuf[lane];
}

// ======================================================================
// Kernel 3b: tiny sequential backward scan over 1024 chunk summaries.
// ======================================================================
__global__ void kf_boundary_kernel(const float* __restrict__ mf_ws,
                                   const float* __restrict__ Pf_ws,
                                   const float* __restrict__ Msum,
                                   const float* __restrict__ csum,
                                   const float* __restrict__ Gsum,
                                   float* __restrict__ bm, float* __restrict__ bP) {
  float m[4], P[16];
#pragma unroll
  for (int i = 0; i < 4; ++i) m[i] = mf_ws[(size_t)(NSTEPS - 1) * 4 + i];
#pragma unroll
  for (int i = 0; i < 16; ++i) P[i] = Pf_ws[(size_t)(NSTEPS - 1) * 16 + i];
  for (int k = NCHUNK - 1; k >= 0; --k) {
#pragma unroll
    for (int i = 0; i < 4; ++i) bm[(size_t)k * 4 + i] = m[i];
#pragma unroll
    for (int i = 0; i < 16; ++i) bP[(size_t)k * 16 + i] = P[i];
    const float* M = &Msum[(size_t)k * 16];
    const float* cv = &csum[(size_t)k * 4];
    const float* G = &Gsum[(size_t)k * 16];
    float mn[4], T[16], Pn[16];
    mat4_vec(M, m, mn);
#pragma unroll
    for (int i = 0; i < 4; ++i) mn[i] += cv[i];
    mat4_mul(M, P, T);
    mat4_mul_nt(T, M, Pn);
#pragma unroll
    for (int i = 0; i < 16; ++i) P[i] = Pn[i] + G[i];
#pragma unroll
    for (int i = 0; i < 4; ++i) m[i] = mn[i];
  }
}

// ======================================================================
// Kernel 3c: per-chunk backward apply; emits means/vars directly.
// ======================================================================
__global__ void kf_apply_kernel(const float* __restrict__ H,
                                const float* __restrict__ J_ws,
                                const float* __restrict__ c_ws,
                                const float* __restrict__ E_ws,
                                const float* __restrict__ bm,
                                const float* __restrict__ bP,
                                float* __restrict__ out) {
  int k = blockIdx.x * blockDim.x + threadIdx.x;
  if (k >= NCHUNK) return;
  float h[4], m[4], P[16];
#pragma unroll
  for (int i = 0; i < 4; ++i) h[i] = H[i];
#pragma unroll
  for (int i = 0; i < 4; ++i) m[i] = bm[(size_t)k * 4 + i];
#pragma unroll
  for (int i = 0; i < 16; ++i) P[i] = bP[(size_t)k * 16 + i];

  for (int n = CHUNK_L - 1; n >= 0; --n) {
    size_t g = (size_t)k * CHUNK_L + n;
    const float* J = &J_ws[g * 16];
    const float* cv = &c_ws[g * 4];
    const float* E = &E_ws[g * 16];
    float mn[4], T[16], Pn[16];
    mat4_vec(J, m, mn);
#pragma unroll
    for (int i = 0; i < 4; ++i) mn[i] += cv[i];
    mat4_mul(J, P, T);
    mat4_mul_nt(T, J, Pn);
#pragma unroll
    for (int i = 0; i < 16; ++i) Pn[i] += E[i];
#pragma unroll
    for (int i = 0; i < 4; ++i)
#pragma unroll
      for (int j = 0; j < 4; ++j)
        P[i * 4 + j] = 0.5f * (Pn[i * 4 + j] + Pn[j * 4 + i]);
#pragma unroll
    for (int i = 0; i < 4; ++i) m[i] = mn[i];

    float mean = 0.f, var = 0.f;
#pragma unroll
    for (int i = 0; i < 4; ++i) {
      mean += h[i] * m[i];
      float rowacc = 0.f;
#pragma unroll
      for (int j = 0; j < 4; ++j) rowacc += P[i * 4 + j] * h[j];
      var += h[i] * rowacc;
    }
    out[g] = mean;
    out[NSTEPS + g] = var;
  }
}

// ======================================================================
extern "C" void kernel_launch(void* const* d_in, const int* in_sizes, int n_in,
                              void* d_out, int out_size, void* d_ws, size_t ws_size,
                              hipStream_t stream) {
  const float* H = (const float*)d_in[1];
  const float* Pinf = (const float*)d_in[2];
  const float* A_seq = (const float*)d_in[3];
  const float* Q_seq = (const float*)d_in[4];
  const float* resid = (const float*)d_in[5];
  const float* maskv = (const float*)d_in[6];
  const float* R_seq = (const float*)d_in[7];
  float* out = (float*)d_out;

  float* ws = (float*)d_ws;
  const size_t N = NSTEPS, C = NCHUNK;
  float* mp_ws = ws;
  float* Pp_ws = mp_ws + 4 * N;
  float* mf_ws = Pp_ws + 16 * N;
  float* Pf_ws = mf_ws + 4 * N;
  float* J_ws = Pf_ws + 16 * N;
  float* c_ws = J_ws + 16 * N;
  float* E_ws = c_ws + 4 * N;
  float* Msum = E_ws + 16 * N;
  float* csum = Msum + 16 * C;
  float* Gsum = csum + 4 * C;
  float* bm = Gsum + 16 * C;
  float* bP = bm + 4 * C;

  kf_filter_kernel<<<1, 32, 0, stream>>>(H, Pinf, A_seq, Q_seq, resid, maskv, R_seq,
                                         mp_ws, Pp_ws, mf_ws, Pf_ws, out);
  kf_prep_kernel<<<NSTEPS / 256, 256, 0, stream>>>(A_seq, mp_ws, Pp_ws, mf_ws, Pf_ws,
                                                   J_ws, c_ws, E_ws);
  kf_summary_kernel<<<NCHUNK / 4, 32, 0, stream>>>(J_ws, c_ws, E_ws, Msum, csum, Gsum);
  kf_boundary_kernel<<<1, 1, 0, stream>>>(mf_ws, Pf_ws, Msum, csum, Gsum, bm, bP);
  kf_apply_kernel<<<NCHUNK / 256, 256, 0, stream>>>(H, J_ws, c_ws, E_ws, bm, bP, out);
}